// PSAR_87574383166108
// MI455X (gfx1250) — compile-verified
//
#include <hip/hip_runtime.h>

// ---------------------------------------------------------------------------
// CDNA5 (gfx1250) implementation of the PSAR patch-routed SR network.
// All conv layers run as implicit-im2col GEMMs on V_WMMA_F32_16X16X4_F32
// (fp32 A/B/C, matches the fp32 reference numerics). The N=3 transposed
// conv runs on VALU. B-fragments are packed pairwise in LDS so each lane
// fetches its two B rows with a single ds_load_b64, and each B fragment is
// reused across 4 (patch) / 2 (MARM) WMMAs to cut LDS traffic per matrix op.
// K3 stages its weight bank with GLOBAL_LOAD_ASYNC_TO_LDS_B128 (ASYNCcnt).
// ---------------------------------------------------------------------------

typedef float v2f __attribute__((ext_vector_type(2)));
typedef float v8f __attribute__((ext_vector_type(8)));

__device__ __forceinline__ v8f wmma_f32(v2f a, v2f b, v8f c) {
  // (neg_a, A, neg_b, B, c_mod, C, reuse_a, reuse_b)
  return __builtin_amdgcn_wmma_f32_16x16x4_f32(false, a, false, b, (short)0, c,
                                               false, false);
}

#define PS_C 1e-4f
#define PM_C 2e-4f
#define PL_C 3e-4f

// d_out layout (floats): y[25165824] | timep[8] | mask[8388608] | y*mask[25165824]
#define T_OFF   25165824
#define M_OFF   25165832
#define YM_OFF  33554440

// ---------------------------------------------------------------------------
// Generic conv stage on a 32x32 patch held in LDS.
// src: LDS [IC<=12][32][32], dst: LDS [12][32][32], sW: LDS paired B-matrix.
// GEMM: M=1024 positions (64 tiles of 16), N=16 (12 used), K=IC*ksz*ksz.
// A layout (16x4 f32): lane&15 = M row; lanes>=16 hold K=k0+2,k0+3.
// B layout (4x16 f32): lane&15 = N col; lanes>=16 hold K rows k0+2,k0+3.
// B pack: sW[(k>>1)*32 + n*2 + (k&1)]  ->  one ds_load_b64 per fragment.
// ---------------------------------------------------------------------------
__device__ __forceinline__ void conv_stage(
    const float* src, float* dst, float* sW,
    const float* __restrict__ wglob, const float* __restrict__ bias,
    int ksz, int K, float slope, bool doPrelu,
    int tid, int lane, int wave)
{
  const int ksq  = ksz * ksz;
  const int pad  = ksz >> 1;
  const int Kpad = (K + 3) & ~3;
  const int half = lane >> 4;
  const int nl   = lane & 15;

  for (int idx = tid; idx < Kpad * 16; idx += 256) {
    int k = idx >> 4, n = idx & 15;
    float v = 0.f;
    if (k < K && n < 12) v = wglob[n * K + k];   // OIHW flatten == oc*K + k
    sW[(k >> 1) * 32 + n * 2 + (k & 1)] = v;
  }
  __syncthreads();

  const int nkk = Kpad >> 2;
  const float bn = (nl < 12) ? bias[nl] : 0.f;

  for (int g = 0; g < 2; ++g) {
    // 4 M-tiles per group: j = g*4+t, tile = wave + 8*j
    int y[4], xp[4], mt[4];
    v8f acc[4];
    #pragma unroll
    for (int t = 0; t < 4; ++t) {
      mt[t] = wave + 8 * (g * 4 + t);
      int mg = mt[t] * 16 + nl;
      y[t] = mg >> 5; xp[t] = mg & 31;
      #pragma unroll
      for (int i = 0; i < 8; ++i) acc[t][i] = bn;
    }
    for (int kk = 0; kk < nkk; ++kk) {
      int k0 = kk * 4 + 2 * half;
      v2f bv = *(const v2f*)(sW + (kk * 2 + half) * 32 + nl * 2);
      int icb[2], dy[2], dx[2]; bool ok[2];
      #pragma unroll
      for (int j = 0; j < 2; ++j) {
        int k = k0 + j;
        ok[j] = (k < K);
        int ic = k / ksq; int t2 = k - ic * ksq;
        int ky = t2 / ksz; int kx = t2 - ky * ksz;
        icb[j] = ic * 1024;
        dy[j] = ky - pad; dx[j] = kx - pad;
      }
      #pragma unroll
      for (int t = 0; t < 4; ++t) {
        v2f av;
        #pragma unroll
        for (int j = 0; j < 2; ++j) {
          int yy = y[t] + dy[j], xc = xp[t] + dx[j];
          float v = 0.f;
          if (ok[j] && (unsigned)yy < 32u && (unsigned)xc < 32u)
            v = src[icb[j] + yy * 32 + xc];
          av[j] = v;
        }
        acc[t] = wmma_f32(av, bv, acc[t]);
      }
    }
    #pragma unroll
    for (int t = 0; t < 4; ++t) {
      if (nl < 12) {
        #pragma unroll
        for (int r = 0; r < 8; ++r) {
          int mgl = mt[t] * 16 + r + 8 * half;  // C/D: VGPR r -> row r (+8 hi)
          int oy = mgl >> 5, ox = mgl & 31;
          float v = acc[t][r];
          if (doPrelu) v = v > 0.f ? v : slope * v;
          dst[nl * 1024 + oy * 32 + ox] = v;
        }
      }
    }
  }
  __syncthreads();
}

// ---------------------------------------------------------------------------
// K1: per-patch pipeline. 512 blocks (one patch each), 256 threads = 8 waves.
// ---------------------------------------------------------------------------
__global__ __launch_bounds__(256) void k1_patch(
    const float* __restrict__ x,
    const float* __restrict__ iicm_w, const float* __restrict__ iicm_b,
    const float* __restrict__ iicm_a,
    const float* __restrict__ w41, const float* __restrict__ b41,
    const float* __restrict__ a41,
    const float* __restrict__ w42, const float* __restrict__ b42,
    const float* __restrict__ w5,  const float* __restrict__ b5,
    const float* __restrict__ w61, const float* __restrict__ b61,
    const float* __restrict__ w62, const float* __restrict__ b62,
    const float* __restrict__ w7,  const float* __restrict__ b7,
    const float* __restrict__ a7,
    const float* __restrict__ pre_a,
    const float* __restrict__ tlp, const float* __restrict__ trp,
    float* __restrict__ label, float* __restrict__ wsTime)
{
  __shared__ float sIn[3 * 34 * 34];               // input patch + 1-halo
  __shared__ __align__(16) float sA[12288];        // 12x32x32 feature ping
  __shared__ __align__(16) float sB[12288];        // 12x32x32 feature pong
  __shared__ __align__(16) float sW[300 * 16];     // paired B-matrix (max K=300)
  __shared__ float sRed[256];
  __shared__ int   sBranch;

  const int tid  = threadIdx.x;
  const int lane = tid & 31;
  const int wave = tid >> 5;
  const int half = lane >> 4;
  const int nl   = lane & 15;

  const int patch = blockIdx.x;
  const int b  = patch >> 6;
  const int pr = (patch >> 3) & 7;
  const int pc = patch & 7;

  for (int idx = tid; idx < 3 * 34 * 34; idx += 256) {
    int ic = idx / 1156; int r = idx - ic * 1156;
    int yy = r / 34, xx = r - (r / 34) * 34;
    int gy = pr * 32 + yy - 1, gx = pc * 32 + xx - 1;
    float v = 0.f;
    if ((unsigned)gy < 256u && (unsigned)gx < 256u)
      v = x[((b * 3 + ic) * 256 + gy) * 256 + gx];
    sIn[idx] = v;
  }
  for (int idx = tid; idx < 28 * 16; idx += 256) { // iicm K=27 -> pad 28, paired
    int k = idx >> 4, n = idx & 15;
    float v = (k < 27 && n < 12) ? iicm_w[n * 27 + k] : 0.f;
    sW[(k >> 1) * 32 + n * 2 + (k & 1)] = v;
  }
  __syncthreads();

  // --- IICM conv (3->12, k3, pad1); halo tile needs no spatial bounds check
  {
    const float aii = iicm_a[0];
    const float bn = (nl < 12) ? iicm_b[nl] : 0.f;
    for (int g = 0; g < 2; ++g) {
      int y[4], xp[4], mt[4];
      v8f acc[4];
      #pragma unroll
      for (int t = 0; t < 4; ++t) {
        mt[t] = wave + 8 * (g * 4 + t);
        int mg = mt[t] * 16 + nl;
        y[t] = mg >> 5; xp[t] = mg & 31;
        #pragma unroll
        for (int i = 0; i < 8; ++i) acc[t][i] = bn;
      }
      for (int kk = 0; kk < 7; ++kk) {
        int k0 = kk * 4 + 2 * half;
        v2f bv = *(const v2f*)(sW + (kk * 2 + half) * 32 + nl * 2);
        int off[2]; bool ok[2];
        #pragma unroll
        for (int j = 0; j < 2; ++j) {
          int k = k0 + j;
          ok[j] = (k < 27);
          int ic = k / 9; int t2 = k - ic * 9;
          off[j] = ic * 1156 + (t2 / 3) * 34 + (t2 - (t2 / 3) * 3);
        }
        #pragma unroll
        for (int t = 0; t < 4; ++t) {
          v2f av;
          #pragma unroll
          for (int j = 0; j < 2; ++j)
            av[j] = ok[j] ? sIn[off[j] + y[t] * 34 + xp[t]] : 0.f;
          acc[t] = wmma_f32(av, bv, acc[t]);
        }
      }
      #pragma unroll
      for (int t = 0; t < 4; ++t) {
        if (nl < 12) {
          #pragma unroll
          for (int r = 0; r < 8; ++r) {
            int mgl = mt[t] * 16 + r + 8 * half;
            int oy = mgl >> 5, ox = mgl & 31;
            float v = acc[t][r];
            v = v > 0.f ? v : aii * v;           // prelu(iicm_a)
            sA[nl * 1024 + oy * 32 + ox] = v;
          }
        }
      }
    }
  }
  __syncthreads();

  // --- ssza = 2*mean(|f|) ; per-patch branch selection (block-uniform)
  {
    float s = 0.f;
    for (int i = tid; i < 12288; i += 256) s += fabsf(sA[i]);
    sRed[tid] = s;
    __syncthreads();
    for (int off = 128; off > 0; off >>= 1) {
      if (tid < off) sRed[tid] += sRed[tid + off];
      __syncthreads();
    }
    if (tid == 0) {
      float ssza = 2.f * sRed[0] / 12288.f;
      float tl = tlp[0], tr = trp[0], thr = tl + tr;
      int br; float tm;
      if (ssza > thr)       { br = 0; tm = PS_C * (ssza - thr); }
      else if (ssza >= tl)  { br = 1; tm = PM_C * (ssza - tl + thr - ssza); }
      else                  { br = 2; tm = PL_C * (tl - ssza); }
      sBranch = br;
      atomicAdd(&wsTime[b], tm * (1.f / 64.f));  // mean over 64 patches
    }
    __syncthreads();
  }
  const int branch = sBranch;

  // --- branch-pruned conv chain (dense where() in ref == pick one branch)
  conv_stage(sA, sB, sW, w41, b41, 5, 300, a41[0], true,  tid, lane, wave);
  conv_stage(sB, sA, sW, w42, b42, 3, 108, 0.f,    false, tid, lane, wave);
  float* cur = sA;                                     // xs (easy)
  if (branch >= 1) {
    conv_stage(sA, sB, sW, w5, b5, 3, 108, 0.f, false, tid, lane, wave);
    cur = sB;                                          // xm (medium)
  }
  if (branch == 2) {
    conv_stage(sB, sA, sW, w61, b61, 3, 108, 0.f, false, tid, lane, wave);
    conv_stage(sA, sB, sW, w62, b62, 3, 108, 0.f, false, tid, lane, wave);
    cur = sB;                                          // xh (hard)
  }

  // --- conv7: prelu(pre_a) -> 1x1 conv 12->56 -> prelu(a7) -> stitch to label
  const float pa  = pre_a[0];
  const float a7v = a7[0];
  for (int nt = 0; nt < 4; ++nt) {
    for (int idx = tid; idx < 12 * 16; idx += 256) {
      int k = idx >> 4, n = idx & 15;
      int oc = nt * 16 + n;
      float v = (oc < 56) ? w7[oc * 12 + k] : 0.f;
      sW[(k >> 1) * 32 + n * 2 + (k & 1)] = v;
    }
    __syncthreads();
    const int oc0 = nt * 16 + nl;
    const float bn = (oc0 < 56) ? b7[oc0] : 0.f;
    for (int mt = wave; mt < 64; mt += 8) {
      v8f acc;
      #pragma unroll
      for (int i = 0; i < 8; ++i) acc[i] = bn;
      int mg = mt * 16 + nl;
      int y = mg >> 5, xx = mg & 31;
      for (int kk = 0; kk < 3; ++kk) {        // K=12
        int k0 = kk * 4 + 2 * half;
        v2f bv = *(const v2f*)(sW + (kk * 2 + half) * 32 + nl * 2);
        v2f av;
        #pragma unroll
        for (int j = 0; j < 2; ++j) {
          float v = cur[(k0 + j) * 1024 + y * 32 + xx];
          av[j] = v > 0.f ? v : pa * v;       // prelu(pre_a) fused into gather
        }
        acc = wmma_f32(av, bv, acc);
      }
      #pragma unroll
      for (int r = 0; r < 8; ++r) {
        if (oc0 < 56) {
          int mgl = mt * 16 + r + 8 * half;
          int oy = mgl >> 5, ox = mgl & 31;
          float v = acc[r];
          v = v > 0.f ? v : a7v * v;          // prelu(a7)
          label[((b * 56 + oc0) * 256 + (pr * 32 + oy)) * 256 + (pc * 32 + ox)] = v;
        }
      }
    }
    __syncthreads();
  }
}

// ---------------------------------------------------------------------------
// K2: MARM fusion. z = prelu(conv3x3(label,56->56) + conv3x3(x,3->56)).
// One block per 16x16 spatial tile; GEMM K = 56*9 + 3*9 = 531 (pad 532).
// Each wave runs 2 M-subtiles through one K loop so the B fragment (one
// ds_load_b64) feeds two WMMAs. Block 0 also publishes timep.
// ---------------------------------------------------------------------------
__global__ __launch_bounds__(256) void k2_marm(
    const float* __restrict__ label, const float* __restrict__ x,
    const float* __restrict__ marm_wx, const float* __restrict__ marm_bx,
    const float* __restrict__ marm_w,  const float* __restrict__ marm_b,
    const float* __restrict__ marm_a,
    float* __restrict__ z,
    const float* __restrict__ wsTime, float* __restrict__ dout)
{
  __shared__ float sLab[56 * 18 * 18];
  __shared__ float sX[3 * 18 * 18];
  __shared__ __align__(16) float sW2[532 * 16];

  const int tid  = threadIdx.x;
  const int lane = tid & 31;
  const int wave = tid >> 5;
  const int half = lane >> 4;
  const int nl   = lane & 15;

  const int blk  = blockIdx.x;
  const int b    = blk >> 8;
  const int tile = blk & 255;
  const int ty0  = (tile >> 4) * 16, tx0 = (tile & 15) * 16;

  if (blk == 0 && tid < 8) dout[T_OFF + tid] = wsTime[tid];

  for (int idx = tid; idx < 56 * 324; idx += 256) {
    int ic = idx / 324; int r = idx - ic * 324;
    int yy = r / 18, xx = r - (r / 18) * 18;
    int gy = ty0 + yy - 1, gx = tx0 + xx - 1;
    float v = 0.f;
    if ((unsigned)gy < 256u && (unsigned)gx < 256u)
      v = label[((b * 56 + ic) * 256 + gy) * 256 + gx];
    sLab[idx] = v;
  }
  for (int idx = tid; idx < 3 * 324; idx += 256) {
    int ic = idx / 324; int r = idx - ic * 324;
    int yy = r / 18, xx = r - (r / 18) * 18;
    int gy = ty0 + yy - 1, gx = tx0 + xx - 1;
    float v = 0.f;
    if ((unsigned)gy < 256u && (unsigned)gx < 256u)
      v = x[((b * 3 + ic) * 256 + gy) * 256 + gx];
    sX[idx] = v;
  }
  __syncthreads();

  const float ma = marm_a[0];
  for (int nt = 0; nt < 4; ++nt) {
    int ocb = nt * 16;
    for (int idx = tid; idx < 532 * 16; idx += 256) {
      int k = idx >> 4, n = idx & 15;
      int oc = ocb + n;
      float v = 0.f;
      if (oc < 56) {
        if (k < 504)      v = marm_w[oc * 504 + k];
        else if (k < 531) v = marm_wx[oc * 27 + (k - 504)];
      }
      sW2[(k >> 1) * 32 + n * 2 + (k & 1)] = v;
    }
    __syncthreads();

    // subtiles st = wave and wave+8  ->  m1 = m0 + 128  ->  y1 = y0 + 8
    v8f acc0, acc1;
    #pragma unroll
    for (int i = 0; i < 8; ++i) { acc0[i] = 0.f; acc1[i] = 0.f; }
    int m0 = wave * 16 + nl;
    int y0 = m0 >> 4, x0 = m0 & 15;

    for (int kk = 0; kk < 133; ++kk) {        // Kpad = 532
      int k0 = kk * 4 + 2 * half;
      v2f bv = *(const v2f*)(sW2 + (kk * 2 + half) * 32 + nl * 2);
      v2f av0, av1;
      #pragma unroll
      for (int j = 0; j < 2; ++j) {
        int k = k0 + j;
        float v0 = 0.f, v1 = 0.f;
        if (k < 531) {
          const float* sp; int ic, t2;
          if (k < 504) { ic = k / 9; t2 = k - ic * 9; sp = sLab; }
          else { int k2 = k - 504; ic = k2 / 9; t2 = k2 - ic * 9; sp = sX; }
          int o0 = ic * 324 + (t2 / 3) * 18 + (t2 - (t2 / 3) * 3)
                 + y0 * 18 + x0;
          v0 = sp[o0];
          v1 = sp[o0 + 144];                   // (y0+8)*18
        }
        av0[j] = v0; av1[j] = v1;
      }
      acc0 = wmma_f32(av0, bv, acc0);
      acc1 = wmma_f32(av1, bv, acc1);
    }

    int oc = ocb + nl;
    if (oc < 56) {
      float bb = marm_b[oc] + marm_bx[oc];
      #pragma unroll
      for (int r = 0; r < 8; ++r) {
        int ml0 = wave * 16 + r + 8 * half;
        int oy0 = ml0 >> 4, ox0 = ml0 & 15;
        float v = acc0[r] + bb;
        v = v > 0.f ? v : ma * v;
        z[((b * 56 + oc) * 256 + (ty0 + oy0)) * 256 + (tx0 + ox0)] = v;
        int ml1 = ml0 + 128;                   // st = wave+8
        int oy1 = ml1 >> 4, ox1 = ml1 & 15;
        float w = acc1[r] + bb;
        w = w > 0.f ? w : ma * w;
        z[((b * 56 + oc) * 256 + (ty0 + oy1)) * 256 + (tx0 + ox1)] = w;
      }
    }
    __syncthreads();
  }
}

// ---------------------------------------------------------------------------
// K3: ConvTranspose2d(56->3, k9, s4, pad4, outpad3) == lhs-dilated conv with
// flipped kernel, pads (4,7). N=3 -> VALU. One block per 64x64 output tile.
// Weight bank staged with the CDNA5 async memory->LDS DMA path (ASYNCcnt).
// ---------------------------------------------------------------------------
__global__ __launch_bounds__(256) void k3_up(
    const float* __restrict__ z, const float* __restrict__ up_w,
    const float* __restrict__ up_b, float* __restrict__ dout)
{
  __shared__ float sZ[56 * 18 * 18];
  __shared__ __align__(16) float sWU[3 * 56 * 81];   // 13608 floats = 3402 x 16B

  const int tid  = threadIdx.x;
  const int blk  = blockIdx.x;
  const int b    = blk >> 8;
  const int tile = blk & 255;
  const int oy0  = (tile >> 4) * 64, ox0 = (tile & 15) * 64;
  const int iy0  = (oy0 >> 2) - 1,   ix0 = (ox0 >> 2) - 1;

  // async DMA: global -> LDS, 16B per lane-op, no VGPR round trip
  for (int idx = tid; idx < 3402; idx += 256) {
    const float* g = up_w + idx * 4;
    unsigned l = (unsigned)(unsigned long long)&sWU[idx * 4];
    asm volatile("global_load_async_to_lds_b128 %0, %1, off"
                 :: "v"(l), "v"(g) : "memory");
  }

  for (int idx = tid; idx < 56 * 324; idx += 256) {
    int ic = idx / 324; int r = idx - ic * 324;
    int yy = r / 18, xx = r - (r / 18) * 18;
    int iy = iy0 + yy, ix = ix0 + xx;
    float v = 0.f;
    if ((unsigned)iy < 256u && (unsigned)ix < 256u)
      v = z[((b * 56 + ic) * 256 + iy) * 256 + ix];
    sZ[idx] = v;
  }

  asm volatile("s_wait_asynccnt 0x0" ::: "memory");
  __syncthreads();

  const float ub0 = up_b[0], ub1 = up_b[1], ub2 = up_b[2];
  for (int s = 0; s < 16; ++s) {
    int p  = s * 256 + tid;
    int oy = oy0 + (p >> 6), ox = ox0 + (p & 63);
    float a0 = ub0, a1 = ub1, a2 = ub2;
    int ry = (4 - (oy & 3)) & 3;                 // tap phase: ky ≡ -oy (mod 4)
    int rx = (4 - (ox & 3)) & 3;
    for (int ky = ry; ky < 9; ky += 4) {
      int iy = (oy + ky - 4) >> 2;
      if ((unsigned)iy >= 256u) continue;
      int liy = iy - iy0;
      for (int kx = rx; kx < 9; kx += 4) {
        int ix = (ox + kx - 4) >> 2;
        if ((unsigned)ix >= 256u) continue;
        int lix = ix - ix0;
        int wb = (8 - ky) * 9 + (8 - kx);        // kernel flip
        const float* zp = &sZ[liy * 18 + lix];
        const float* wp = &sWU[wb];
        #pragma unroll 8
        for (int ic = 0; ic < 56; ++ic) {
          float v = zp[ic * 324];
          a0 += wp[ic * 81] * v;
          a1 += wp[4536 + ic * 81] * v;
          a2 += wp[9072 + ic * 81] * v;
        }
      }
    }
    int i0 = ((b * 3 + 0) * 1024 + oy) * 1024 + ox;
    int i1 = i0 + 1048576;
    int i2 = i1 + 1048576;
    dout[i0] = a0; dout[i1] = a1; dout[i2] = a2;                 // y
    dout[YM_OFF + i0] = a0; dout[YM_OFF + i1] = a1; dout[YM_OFF + i2] = a2;
    dout[M_OFF + b * 1048576 + oy * 1024 + ox] = 1.0f;           // mask
  }
}

// ---------------------------------------------------------------------------
extern "C" void kernel_launch(void* const* d_in, const int* in_sizes, int n_in,
                              void* d_out, int out_size, void* d_ws,
                              size_t ws_size, hipStream_t stream)
{
  (void)in_sizes; (void)n_in; (void)out_size; (void)ws_size;

  const float* x       = (const float*)d_in[0];
  const float* iicm_w  = (const float*)d_in[1];
  const float* iicm_b  = (const float*)d_in[2];
  const float* iicm_a  = (const float*)d_in[3];
  const float* w41     = (const float*)d_in[4];
  const float* b41     = (const float*)d_in[5];
  const float* a41     = (const float*)d_in[6];
  const float* w42     = (const float*)d_in[7];
  const float* b42     = (const float*)d_in[8];
  const float* w5      = (const float*)d_in[9];
  const float* b5      = (const float*)d_in[10];
  const float* w61     = (const float*)d_in[11];
  const float* b61     = (const float*)d_in[12];
  const float* w62     = (const float*)d_in[13];
  const float* b62     = (const float*)d_in[14];
  const float* w7      = (const float*)d_in[15];
  const float* b7      = (const float*)d_in[16];
  const float* a7      = (const float*)d_in[17];
  const float* pre_a   = (const float*)d_in[18];
  const float* marm_wx = (const float*)d_in[19];
  const float* marm_bx = (const float*)d_in[20];
  const float* marm_w  = (const float*)d_in[21];
  const float* marm_b  = (const float*)d_in[22];
  const float* marm_a  = (const float*)d_in[23];
  const float* up_w    = (const float*)d_in[24];
  const float* up_b    = (const float*)d_in[25];
  const float* tl      = (const float*)d_in[26];
  const float* tr      = (const float*)d_in[27];

  float* dout   = (float*)d_out;
  float* ws     = (float*)d_ws;
  float* label  = ws;                       // 8*56*256*256 = 29360128 floats
  float* zbuf   = ws + 29360128;            // same size
  float* wsTime = ws + 2 * 29360128;        // 8 floats

  hipMemsetAsync(wsTime, 0, 8 * sizeof(float), stream);

  k1_patch<<<512, 256, 0, stream>>>(
      x, iicm_w, iicm_b, iicm_a, w41, b41, a41, w42, b42, w5, b5,
      w61, b61, w62, b62, w7, b7, a7, pre_a, tl, tr, label, wsTime);

  k2_marm<<<2048, 256, 0, stream>>>(
      label, x, marm_wx, marm_bx, marm_w, marm_b, marm_a, zbuf, wsTime, dout);

  k3_up<<<2048, 256, 0, stream>>>(zbuf, up_w, up_b, dout);
}